// GATModel_67740224192530
// MI455X (gfx1250) — compile-verified
//
#include <hip/hip_runtime.h>
#include <hip/hip_bf16.h>

#define NN 50000
#define NE 500000
#define EF (NE + NN)        // 550000 edges incl. self loops
#define NG 512
#define FIN 128
#define FE 16
#define HID 64
#define HEADS 4
#define FOUT 256            // HEADS*HID

typedef float v2f __attribute__((ext_vector_type(2)));
typedef float v8f __attribute__((ext_vector_type(8)));

// ---------- helpers ----------
__device__ __forceinline__ unsigned fenc(float f) {
  unsigned u = __float_as_uint(f);
  return (u & 0x80000000u) ? ~u : (u | 0x80000000u);
}
__device__ __forceinline__ float fdec(unsigned k) {
  return (k & 0x80000000u) ? __uint_as_float(k & 0x7FFFFFFFu) : __uint_as_float(~k);
}

__global__ void fill_f32(float* __restrict__ p, float v, long n) {
  long i = (long)blockIdx.x * blockDim.x + threadIdx.x;
  long st = (long)gridDim.x * blockDim.x;
  for (; i < n; i += st) p[i] = v;
}
__global__ void fill_u32(unsigned* __restrict__ p, unsigned v, long n) {
  long i = (long)blockIdx.x * blockDim.x + threadIdx.x;
  long st = (long)gridDim.x * blockDim.x;
  for (; i < n; i += st) p[i] = v;
}

// ---------- self-loop attr: deg + sums + divide ----------
__global__ void deg_kernel(const int* __restrict__ ei, float* __restrict__ deg) {
  int e = blockIdx.x * blockDim.x + threadIdx.x;
  if (e >= NE) return;
  atomicAdd(&deg[ei[NE + e]], 1.0f);
}
__global__ void loopsum_kernel(const int* __restrict__ ei, const float* __restrict__ eattr,
                               float* __restrict__ loopA) {
  long t = (long)blockIdx.x * blockDim.x + threadIdx.x;
  if (t >= (long)NE * FE) return;
  int e = (int)(t >> 4), f = (int)(t & 15);
  atomicAdd(&loopA[(size_t)ei[NE + e] * FE + f], eattr[t]);
}
__global__ void loopdiv_kernel(float* __restrict__ loopA, const float* __restrict__ deg) {
  long t = (long)blockIdx.x * blockDim.x + threadIdx.x;
  if (t >= (long)NN * FE) return;
  loopA[t] /= fmaxf(deg[t >> 4], 1.0f);
}

// ---------- WMMA fp32 GEMM: C[M,256] = A[M,K] @ B[K,256], M=50000 ----------
// block = 128 threads (4 waves); wave w -> 16 rows x 64 cols (4 accumulators)
__global__ void __launch_bounds__(128)
gemm_f32_wmma(const float* __restrict__ A, const float* __restrict__ B,
              float* __restrict__ C, int K) {
  const int wave = threadIdx.x >> 5;
  const int lane = threadIdx.x & 31;
  const int lmod = lane & 15;
  const int lhi  = lane >> 4;              // 0 or 1
  const int row0 = blockIdx.x * 16;
  const int col0 = wave * 64;

  v8f acc0 = {}; v8f acc1 = {}; v8f acc2 = {}; v8f acc3 = {};

  const float* arow = A + (size_t)(row0 + lmod) * K;
  for (int k = 0; k < K; k += 4) {
    // A frag: lanes 0-15 rows, VGPR0/1 = K0/K1; lanes 16-31 = K2/K3
    v2f a = *(const v2f*)(arow + k + lhi * 2);
    const float* bp0 = B + (size_t)(k + lhi * 2) * FOUT + col0 + lmod;
    const float* bp1 = bp0 + FOUT;
    v2f b;
    b.x = bp0[0];  b.y = bp1[0];
    acc0 = __builtin_amdgcn_wmma_f32_16x16x4_f32(false, a, false, b, (short)0, acc0, false, false);
    b.x = bp0[16]; b.y = bp1[16];
    acc1 = __builtin_amdgcn_wmma_f32_16x16x4_f32(false, a, false, b, (short)0, acc1, false, false);
    b.x = bp0[32]; b.y = bp1[32];
    acc2 = __builtin_amdgcn_wmma_f32_16x16x4_f32(false, a, false, b, (short)0, acc2, false, false);
    b.x = bp0[48]; b.y = bp1[48];
    acc3 = __builtin_amdgcn_wmma_f32_16x16x4_f32(false, a, false, b, (short)0, acc3, false, false);
  }
  // D layout: VGPR v -> row = v (lanes 0-15) / v+8 (lanes 16-31), col = lane%16
  #pragma unroll
  for (int v = 0; v < 8; ++v) {
    float* crow = C + (size_t)(row0 + v + lhi * 8) * FOUT + col0 + lmod;
    crow[0]  = acc0[v];
    crow[16] = acc1[v];
    crow[32] = acc2[v];
    crow[48] = acc3[v];
  }
}

// ---------- per-node attention dots ----------
__global__ void node_dots(const float* __restrict__ h, const float* __restrict__ a_src,
                          const float* __restrict__ a_dst,
                          float* __restrict__ asrc, float* __restrict__ adst) {
  int t = blockIdx.x * blockDim.x + threadIdx.x;
  if (t >= NN * HEADS) return;
  int n = t >> 2, hd = t & 3;
  const float* hp = h + (size_t)n * FOUT + hd * HID;
  const float* sp = a_src + hd * HID;
  const float* dp = a_dst + hd * HID;
  float s1 = 0.f, s2 = 0.f;
  #pragma unroll 8
  for (int c = 0; c < HID; ++c) { float hv = hp[c]; s1 += hv * sp[c]; s2 += hv * dp[c]; }
  asrc[t] = s1; adst[t] = s2;
}

// ---------- ve[f,head] = sum_c We[f, head*64+c] * a_e[head,c] ----------
__global__ void compute_ve(const float* __restrict__ We, const float* __restrict__ a_e,
                           float* __restrict__ ve) {
  int t = threadIdx.x;
  if (t >= FE * HEADS) return;
  int f = t >> 2, hd = t & 3;
  float s = 0.f;
  for (int c = 0; c < HID; ++c) s += We[(size_t)f * FOUT + hd * HID + c] * a_e[hd * HID + c];
  ve[f * 4 + hd] = s;
}

// ---------- edge pass 1: alpha + leaky relu + segment max ----------
__global__ void __launch_bounds__(256)
edge_pass1(const int* __restrict__ ei, const float* __restrict__ eattr,
           const float* __restrict__ loopA, const float* __restrict__ ve,
           const float* __restrict__ asrc, const float* __restrict__ adst,
           float* __restrict__ alpha, unsigned* __restrict__ amax) {
  int ef = blockIdx.x * blockDim.x + threadIdx.x;
  if (ef >= EF) return;
  int s, d; const float* ea;
  if (ef < NE) { s = ei[ef]; d = ei[NE + ef]; ea = eattr + (size_t)ef * FE; }
  else { s = ef - NE; d = s; ea = loopA + (size_t)s * FE; }
  float ae[4] = {0.f, 0.f, 0.f, 0.f};
  #pragma unroll
  for (int f = 0; f < FE; ++f) {
    float ev = ea[f];
    ae[0] += ev * ve[f * 4 + 0];
    ae[1] += ev * ve[f * 4 + 1];
    ae[2] += ev * ve[f * 4 + 2];
    ae[3] += ev * ve[f * 4 + 3];
  }
  #pragma unroll
  for (int hd = 0; hd < HEADS; ++hd) {
    float al = asrc[s * 4 + hd] + adst[d * 4 + hd] + ae[hd];
    al = al > 0.f ? al : 0.2f * al;                // leaky_relu(0.2)
    alpha[(size_t)ef * 4 + hd] = al;
    atomicMax(&amax[d * 4 + hd], fenc(al));
  }
}

// ---------- edge pass 2: w = exp(alpha - max); denom += w (overwrite alpha) ----------
__global__ void __launch_bounds__(256)
edge_pass2(const int* __restrict__ ei, float* __restrict__ alpha,
           const unsigned* __restrict__ amax, float* __restrict__ denom) {
  int ef = blockIdx.x * blockDim.x + threadIdx.x;
  if (ef >= EF) return;
  int d = (ef < NE) ? ei[NE + ef] : (ef - NE);
  #pragma unroll
  for (int hd = 0; hd < HEADS; ++hd) {
    float w = expf(alpha[(size_t)ef * 4 + hd] - fdec(amax[d * 4 + hd]));
    alpha[(size_t)ef * 4 + hd] = w;
    atomicAdd(&denom[d * 4 + hd], w);
  }
}

// ---------- edge pass 3: out[dst] += attn * h[src] (one wave per edge) ----------
__global__ void __launch_bounds__(256)
edge_pass3(const int* __restrict__ ei, const float* __restrict__ wbuf,
           const float* __restrict__ denom, const float* __restrict__ h,
           float* __restrict__ out) {
  int warp = threadIdx.x >> 5;
  int lane = threadIdx.x & 31;
  long ef = (long)blockIdx.x * 8 + warp;
  if (ef >= EF) return;
  int s, d;
  if (ef < NE) { s = ei[ef]; d = ei[NE + ef]; }
  else { s = (int)(ef - NE); d = s; }
  int hd = lane >> 3;                              // 8 lanes per head (64 ch)
  float w = wbuf[ef * 4 + hd];
  float attn = w / (denom[d * 4 + hd] + 1e-16f);
  const float* hp = h + (size_t)s * FOUT + lane * 8;
  float* op = out + (size_t)d * FOUT + lane * 8;
  #pragma unroll
  for (int i = 0; i < 8; ++i) atomicAdd(&op[i], attn * hp[i]);
}

// ---------- finalize: x = elu(acc + bias) ----------
__global__ void finalize_elu(float* __restrict__ x, const float* __restrict__ bias, long n) {
  long i = (long)blockIdx.x * blockDim.x + threadIdx.x;
  if (i >= n) return;
  float v = x[i] + bias[(int)(i & 255)];
  x[i] = v > 0.f ? v : expm1f(v);
}

// ---------- pooling ----------
__global__ void pool_cnt(const int* __restrict__ batch, float* __restrict__ cnt) {
  int n = blockIdx.x * blockDim.x + threadIdx.x;
  if (n >= NN) return;
  atomicAdd(&cnt[batch[n]], 1.0f);
}
__global__ void pool_sum(const float* __restrict__ x, const int* __restrict__ batch,
                         float* __restrict__ pooled) {
  long i = (long)blockIdx.x * blockDim.x + threadIdx.x;
  if (i >= (long)NN * FOUT) return;
  int n = (int)(i >> 8), c = (int)(i & 255);
  atomicAdd(&pooled[(size_t)batch[n] * FOUT + c], x[i]);
}

// ---------- MLP head: per-graph block of 64 threads ----------
__global__ void __launch_bounds__(64)
mlp_head(const float* __restrict__ pooled, const float* __restrict__ cnt,
         const float* __restrict__ w1, const float* __restrict__ b1,
         const float* __restrict__ w2, const float* __restrict__ b2,
         float* __restrict__ out) {
  int g = blockIdx.x, j = threadIdx.x;
  __shared__ float red[64];
  float inv = 1.0f / fmaxf(cnt[g], 1.0f);
  const float* p = pooled + (size_t)g * FOUT;
  float acc = b1[j];
  for (int k = 0; k < FOUT; ++k) acc += p[k] * inv * w1[k * 64 + j];
  acc = fmaxf(acc, 0.f);
  red[j] = acc * w2[j];
  __syncthreads();
  for (int st = 32; st > 0; st >>= 1) { if (j < st) red[j] += red[j + st]; __syncthreads(); }
  if (j == 0) out[g] = red[0] + b2[0];
}

// ---------- host side ----------
static inline void fillf(float* p, float v, long n, hipStream_t s) {
  int blocks = (int)((n + 255) / 256); if (blocks > 4096) blocks = 4096;
  fill_f32<<<blocks, 256, 0, s>>>(p, v, n);
}
static inline void fillu(unsigned* p, unsigned v, long n, hipStream_t s) {
  int blocks = (int)((n + 255) / 256); if (blocks > 4096) blocks = 4096;
  fill_u32<<<blocks, 256, 0, s>>>(p, v, n);
}

extern "C" void kernel_launch(void* const* d_in, const int* in_sizes, int n_in,
                              void* d_out, int out_size, void* d_ws, size_t ws_size,
                              hipStream_t stream) {
  const float* x_in   = (const float*)d_in[0];     // [N,128]
  const int*   ei     = (const int*)d_in[1];       // [2,E]
  const float* eattr  = (const float*)d_in[2];     // [E,16]
  const int*   batch  = (const int*)d_in[3];       // [N]
  const float* w1 = (const float*)d_in[22];
  const float* b1 = (const float*)d_in[23];
  const float* w2 = (const float*)d_in[24];
  const float* b2 = (const float*)d_in[25];

  float* ws = (float*)d_ws;
  float* X      = ws;  ws += (size_t)NN * FOUT;     // layer in/out buffer
  float* H      = ws;  ws += (size_t)NN * FOUT;     // h = x@W
  float* deg    = ws;  ws += NN;
  float* loopA  = ws;  ws += (size_t)NN * FE;
  float* asrc   = ws;  ws += (size_t)NN * HEADS;
  float* adst   = ws;  ws += (size_t)NN * HEADS;
  float* alpha  = ws;  ws += (size_t)EF * HEADS;    // alpha, then w
  unsigned* amax = (unsigned*)ws; ws += (size_t)NN * HEADS;
  float* denom  = ws;  ws += (size_t)NN * HEADS;
  float* ve     = ws;  ws += FE * HEADS;
  float* pooled = ws;  ws += (size_t)NG * FOUT;
  float* cnt    = ws;  ws += NG;

  // self-loop attrs (fill_value='mean')
  fillf(deg, 0.f, NN, stream);
  fillf(loopA, 0.f, (long)NN * FE, stream);
  deg_kernel<<<(NE + 255) / 256, 256, 0, stream>>>(ei, deg);
  loopsum_kernel<<<(int)(((long)NE * FE + 255) / 256), 256, 0, stream>>>(ei, eattr, loopA);
  loopdiv_kernel<<<(int)(((long)NN * FE + 255) / 256), 256, 0, stream>>>(loopA, deg);

  const float* xin = x_in;
  int K = FIN;
  for (int l = 0; l < 3; ++l) {
    const float* W    = (const float*)d_in[4 + 6 * l + 0];
    const float* a_s  = (const float*)d_in[4 + 6 * l + 1];
    const float* a_d  = (const float*)d_in[4 + 6 * l + 2];
    const float* We   = (const float*)d_in[4 + 6 * l + 3];
    const float* a_e  = (const float*)d_in[4 + 6 * l + 4];
    const float* bias = (const float*)d_in[4 + 6 * l + 5];

    gemm_f32_wmma<<<NN / 16, 128, 0, stream>>>(xin, W, H, K);
    node_dots<<<(NN * HEADS + 255) / 256, 256, 0, stream>>>(H, a_s, a_d, asrc, adst);
    compute_ve<<<1, 64, 0, stream>>>(We, a_e, ve);

    fillu(amax, 0x007FFFFFu, (long)NN * HEADS, stream);   // encode(-inf)
    fillf(denom, 0.f, (long)NN * HEADS, stream);
    edge_pass1<<<(EF + 255) / 256, 256, 0, stream>>>(ei, eattr, loopA, ve, asrc, adst, alpha, amax);
    edge_pass2<<<(EF + 255) / 256, 256, 0, stream>>>(ei, alpha, amax, denom);

    fillf(X, 0.f, (long)NN * FOUT, stream);               // xin dead after GEMM
    edge_pass3<<<EF / 8, 256, 0, stream>>>(ei, alpha, denom, H, X);
    finalize_elu<<<(int)(((long)NN * FOUT + 255) / 256), 256, 0, stream>>>(X, bias, (long)NN * FOUT);

    xin = X; K = FOUT;
  }

  // global mean pool + MLP
  fillf(pooled, 0.f, (long)NG * FOUT, stream);
  fillf(cnt, 0.f, NG, stream);
  pool_cnt<<<(NN + 255) / 256, 256, 0, stream>>>(batch, cnt);
  pool_sum<<<(int)(((long)NN * FOUT + 255) / 256), 256, 0, stream>>>(X, batch, pooled);
  mlp_head<<<NG, 64, 0, stream>>>(pooled, cnt, w1, b1, w2, b2, (float*)d_out);
}